// DictionaryLearningGumbelSoftmaxMultiView_38182259262175
// MI455X (gfx1250) — compile-verified
//
#include <hip/hip_runtime.h>
#include <math.h>

typedef __attribute__((ext_vector_type(2))) float v2f;
typedef __attribute__((ext_vector_type(8))) float v8f;

#define Vn 3
#define Cn 15
#define Hn 768
#define NPV 4096
#define Nn (Vn * NPV)
#define NEG (Cn - 1)
#define TEMP_ 0.5f
#define BN_EPS_ 1e-5f
#define GEPS 1e-20f

// ---------------------------------------------------------------------------
// Kernel 1: h = x @ w_t^T + b_t using V_WMMA_F32_16X16X4_F32 (fp32 tensor op)
// Block: 256 threads = 8 waves; block tile 32(M) x 64(N); K-tile 32.
// Wave (waveM, waveN) computes one 16x16 C tile; 8 WMMA per K-tile.
// ---------------------------------------------------------------------------
#define TM 32
#define TN 64
#define TK 32

__global__ __launch_bounds__(256) void k_gemm_h(
    const float* __restrict__ x, const float* __restrict__ w_t,
    const float* __restrict__ b_t, float* __restrict__ h)
{
    __shared__ float As[TM][36];   // [m][k], row stride 36 floats (16B aligned)
    __shared__ float Bs[TN][36];   // [n][k] = w_t[n][k]  (B matrix transposed view)

    const int tid   = threadIdx.x;
    const int lane  = tid & 31;
    const int wv    = tid >> 5;
    const int waveM = wv >> 2;           // 0..1
    const int waveN = wv & 3;            // 0..3
    const int mbase = blockIdx.y * TM;
    const int nbase = blockIdx.x * TN;

    // A-frag (16x4 f32): lane l holds M=l&15, K = 2*(l>>4) + {0,1}  (2 VGPRs)
    // B-frag (4x16 f32): lane l holds N=l&15, K = 2*(l>>4) + {0,1}
    const int lm   = (lane & 15) + waveM * 16;
    const int ln   = (lane & 15) + waveN * 16;
    const int koff = (lane >> 4) * 2;

    v8f acc = {0.f, 0.f, 0.f, 0.f, 0.f, 0.f, 0.f, 0.f};

    for (int k0 = 0; k0 < Hn; k0 += TK) {
        // Stage A tile: 32x32 floats, one float4 per thread
        {
            const int row  = tid >> 3;          // 0..31
            const int col4 = (tid & 7) * 4;     // 0..28
            const float4 a4 =
                *(const float4*)(x + (size_t)(mbase + row) * Hn + k0 + col4);
            *(float4*)&As[row][col4] = a4;
            if (k0 + TK < Hn)  // speculative prefetch of next A tile -> global_prefetch_b8
                __builtin_prefetch(x + (size_t)(mbase + row) * Hn + k0 + TK + col4, 0, 1);
        }
        // Stage B tile: Bs[n][k] = w_t[nbase+n][k0+k], 64x32 floats, two float4/thread
        {
            const int n  = tid >> 2;            // 0..63
            const int kk = (tid & 3) * 4;       // 0..12
            const float* wp = w_t + (size_t)(nbase + n) * Hn + k0;
            const float4 b0 = *(const float4*)(wp + kk);
            const float4 b1 = *(const float4*)(wp + kk + 16);
            *(float4*)&Bs[n][kk]      = b0;
            *(float4*)&Bs[n][kk + 16] = b1;
        }
        __syncthreads();

#pragma unroll
        for (int ks = 0; ks < TK; ks += 4) {
            v2f a = *(const v2f*)&As[lm][ks + koff];
            v2f b = *(const v2f*)&Bs[ln][ks + koff];
            acc = __builtin_amdgcn_wmma_f32_16x16x4_f32(
                false, a, false, b, (short)0, acc, false, false);
        }
        __syncthreads();
    }

    // C/D layout: VGPR r, lane l -> M = r + 8*(l>>4), N = l&15
    const int gcol  = nbase + waveN * 16 + (lane & 15);
    const float bias = b_t[gcol];
    const int rbase = mbase + waveM * 16 + (lane >> 4) * 8;
#pragma unroll
    for (int r = 0; r < 8; ++r)
        h[(size_t)(rbase + r) * Hn + gcol] = acc[r] + bias;
}

// ---------------------------------------------------------------------------
// Kernel 2/3: deterministic two-stage per-column mean/var for BatchNorm
// ---------------------------------------------------------------------------
#define RCHUNKS 32
#define RPC (Nn / RCHUNKS)   // 384 rows per chunk

__global__ __launch_bounds__(256) void k_stats(
    const float* __restrict__ h, float* __restrict__ psum, float* __restrict__ psq)
{
    const int col   = blockIdx.x * 256 + threadIdx.x;   // 0..767
    const int rbase = blockIdx.y * RPC;
    float s = 0.f, q = 0.f;
    for (int r = 0; r < RPC; ++r) {
        float v = h[(size_t)(rbase + r) * Hn + col];
        s += v; q += v * v;
    }
    psum[blockIdx.y * Hn + col] = s;
    psq [blockIdx.y * Hn + col] = q;
}

__global__ __launch_bounds__(256) void k_finalize(
    const float* __restrict__ psum, const float* __restrict__ psq,
    const float* __restrict__ gamma, const float* __restrict__ beta,
    float* __restrict__ scale, float* __restrict__ shift)
{
    const int col = blockIdx.x * 256 + threadIdx.x;
    float s = 0.f, q = 0.f;
    for (int i = 0; i < RCHUNKS; ++i) { s += psum[i * Hn + col]; q += psq[i * Hn + col]; }
    const float mu  = s / (float)Nn;
    const float var = q / (float)Nn - mu * mu;
    const float rs  = rsqrtf(var + BN_EPS_);
    const float sc  = gamma[col] * rs;
    scale[col] = sc;
    shift[col] = beta[col] - mu * sc;
}

// ---------------------------------------------------------------------------
// Kernel 4: fused BN+ReLU -> logits -> softmax/Gumbel/top-1 -> outputs
// 16 rows per block; dicts view cached in LDS (46 KB of the 320 KB WGP pool).
// Dominant cost: 14 coalesced dictionary-row copies per row (528 MB stores).
// ---------------------------------------------------------------------------
#define RPB 16

__global__ __launch_bounds__(256) void k_fused(
    const float* __restrict__ h, const float* __restrict__ scale,
    const float* __restrict__ shift, const float* __restrict__ u,
    const float* __restrict__ w_proj, const float* __restrict__ b_proj,
    const float* __restrict__ dicts,
    float* __restrict__ o_gw, float* __restrict__ o_recon,
    float* __restrict__ o_logits, float* __restrict__ o_neg,
    float* __restrict__ o_revp)
{
    __shared__ float dictS[Cn * Hn];   // 46080 B
    __shared__ float hrow[Hn];         //  3072 B
    __shared__ float wred[8][Cn];
    __shared__ float logitS[Cn];
    __shared__ float gwS[Cn];
    __shared__ int   negS[NEG];

    const int tid  = threadIdx.x;
    const int lane = tid & 31;
    const int wid  = tid >> 5;
    const int v    = blockIdx.x / (NPV / RPB);

    for (int i = tid; i < Cn * Hn; i += 256)
        dictS[i] = dicts[(size_t)v * Cn * Hn + i];
    __syncthreads();

    for (int rr = 0; rr < RPB; ++rr) {
        const int row = blockIdx.x * RPB + rr;

        // BN + ReLU
        for (int c = tid; c < Hn; c += 256) {
            float hv = h[(size_t)row * Hn + c] * scale[c] + shift[c];
            hrow[c] = fmaxf(hv, 0.f);
        }
        __syncthreads();

        // logits: 15 dot-products over H=768 (wave32 shuffle reduction)
        float acc[Cn];
#pragma unroll
        for (int c = 0; c < Cn; ++c) acc[c] = 0.f;
#pragma unroll
        for (int j = 0; j < 3; ++j) {
            const int col = tid + j * 256;
            const float hv = hrow[col];
#pragma unroll
            for (int c = 0; c < Cn; ++c)
                acc[c] += hv * w_proj[((size_t)v * Cn + c) * Hn + col];
        }
#pragma unroll
        for (int c = 0; c < Cn; ++c) {
            float s = acc[c];
#pragma unroll
            for (int off = 16; off > 0; off >>= 1)
                s += __shfl_xor(s, off, 32);
            if (lane == 0) wred[wid][c] = s;
        }
        __syncthreads();
        if (tid < Cn) {
            float s = 0.f;
            for (int w = 0; w < 8; ++w) s += wred[w][tid];
            logitS[tid] = s + b_proj[v * Cn + tid];
        }
        __syncthreads();

        // Per-row scalar pipeline (15 wide -> trivial): softmax, gumbel, top-1
        if (tid == 0) {
            float lg[Cn], p[Cn], g[Cn];
            float mx = -3.4e38f;
            for (int c = 0; c < Cn; ++c) { lg[c] = logitS[c]; mx = fmaxf(mx, lg[c]); }
            float se = 0.f;
            for (int c = 0; c < Cn; ++c) { p[c] = __expf(lg[c] - mx); se += p[c]; }
            const float inv = 1.f / se;
            for (int c = 0; c < Cn; ++c) p[c] *= inv;

            float mg = -3.4e38f;
            for (int c = 0; c < Cn; ++c) {
                const float uu = u[(size_t)row * Cn + c];
                const float gn = -__logf(-__logf(uu + GEPS) + GEPS);
                g[c] = (lg[c] + gn) * (1.f / TEMP_);
                mg = fmaxf(mg, g[c]);
            }
            float sg = 0.f;
            for (int c = 0; c < Cn; ++c) { g[c] = __expf(g[c] - mg); sg += g[c]; }
            const float invg = 1.f / sg;
            int am = 0; float best = -1.f;
            for (int c = 0; c < Cn; ++c) {
                g[c] *= invg;
                if (g[c] > best) { best = g[c]; am = c; }   // first max (tie rule)
                gwS[c] = g[c];
                o_gw    [(size_t)row * Cn + c] = g[c];
                o_logits[(size_t)row * Cn + c] = lg[c];
            }
            // negatives (K=1 -> all classes except argmax, ascending) + rev softmax
            float rg[NEG]; float mr = -3.4e38f; int j2 = 0;
            for (int c = 0; c < Cn; ++c) {
                if (c == am) continue;
                negS[j2] = c;
                float rv = 1.f - p[c]; rv *= rv;
                rg[j2] = rv; mr = fmaxf(mr, rv); ++j2;
            }
            float sr = 0.f;
            for (int j = 0; j < NEG; ++j) { rg[j] = __expf(rg[j] - mr); sr += rg[j]; }
            const float invr = 1.f / sr;
            for (int j = 0; j < NEG; ++j)
                o_revp[(size_t)row * NEG + j] = rg[j] * invr;
        }
        __syncthreads();

        // recon = gw . dicts  and the bandwidth-dominant neg_samples copies
#pragma unroll
        for (int j = 0; j < 3; ++j) {
            const int col = tid + j * 256;
            float r = 0.f;
#pragma unroll
            for (int c = 0; c < Cn; ++c) r += gwS[c] * dictS[c * Hn + col];
            o_recon[(size_t)row * Hn + col] = r;
        }
        for (int nj = 0; nj < NEG; ++nj) {
            const int nc = negS[nj];
            const size_t ob = ((size_t)row * NEG + nj) * Hn;
#pragma unroll
            for (int j = 0; j < 3; ++j) {
                const int col = tid + j * 256;
                o_neg[ob + col] = dictS[nc * Hn + col];
            }
        }
        __syncthreads();
    }
}

// ---------------------------------------------------------------------------
extern "C" void kernel_launch(void* const* d_in, const int* in_sizes, int n_in,
                              void* d_out, int out_size, void* d_ws, size_t ws_size,
                              hipStream_t stream)
{
    const float* x      = (const float*)d_in[0];
    const float* u      = (const float*)d_in[1];
    const float* w_t    = (const float*)d_in[2];
    const float* b_t    = (const float*)d_in[3];
    const float* gamma  = (const float*)d_in[4];
    const float* beta   = (const float*)d_in[5];
    const float* w_proj = (const float*)d_in[6];
    const float* b_proj = (const float*)d_in[7];
    const float* dicts  = (const float*)d_in[8];

    // workspace layout (floats): h | psum | psq | scale | shift  (~36 MB)
    float* h     = (float*)d_ws;
    float* psum  = h + (size_t)Nn * Hn;
    float* psq   = psum + RCHUNKS * Hn;
    float* scale = psq + RCHUNKS * Hn;
    float* shift = scale + Hn;

    // output layout: gw | recon | logits | neg_samples | rev_probs
    float* out      = (float*)d_out;
    float* o_gw     = out;
    float* o_recon  = o_gw + (size_t)Nn * Cn;
    float* o_logits = o_recon + (size_t)Nn * Hn;
    float* o_neg    = o_logits + (size_t)Nn * Cn;
    float* o_revp   = o_neg + (size_t)Nn * NEG * Hn;

    dim3 g1(Hn / TN, Nn / TM);            // 12 x 384 blocks
    k_gemm_h<<<g1, 256, 0, stream>>>(x, w_t, b_t, h);

    dim3 g2(Hn / 256, RCHUNKS);           // 3 x 32 blocks
    k_stats<<<g2, 256, 0, stream>>>(h, psum, psq);

    k_finalize<<<Hn / 256, 256, 0, stream>>>(psum, psq, gamma, beta, scale, shift);

    k_fused<<<Nn / RPB, 256, 0, stream>>>(h, scale, shift, u, w_proj, b_proj, dicts,
                                          o_gw, o_recon, o_logits, o_neg, o_revp);
}